// Norm_25795573580460
// MI455X (gfx1250) — compile-verified
//
#include <hip/hip_runtime.h>

typedef __attribute__((ext_vector_type(2))) float v2f;
typedef __attribute__((ext_vector_type(8))) float v8f;

#define LROWS 2048
#define NBATCH 8192
#define EPSF 1e-5f
#define TILE 256                 // rows per LDS tile
#define NTILES (LROWS / TILE)    // 8
#define PADW 16                  // padded LDS row width (floats)

// ---------------------------------------------------------------------------
// Kernel 1: per-batch reduction via V_WMMA_F32_16X16X4_F32.
// One block (256 thr = 8 waves) per batch.
// LDS tiles are padded to 16 cols: cols 0..2 = data, col 3 = 1.0 (ones
// row/col so one WMMA chain produces sum(proj_k*x_j), sum(x_j), sum(proj_k)
// simultaneously), cols 4..15 = 0. Fragments are then plain ds_loads:
//   A[m][K] = ldsP[row(K)*16 + m],  B[K][n] = ldsX[row(K)*16 + n]
// Double-buffered tiles overlap next tile's global loads with WMMA compute.
// ---------------------------------------------------------------------------
__global__ __launch_bounds__(256) void norm_reduce_kernel(
    const float* __restrict__ x, const float* __restrict__ g,
    const float* __restrict__ bvec, const float* __restrict__ proj,
    float* __restrict__ params) {
  __shared__ float smem[2][2][TILE * PADW];  // [buf][P=0/X=1][row*16+col], 64 KB
  __shared__ float part[8 * 16];

  const int batch = blockIdx.x;
  const int tid   = threadIdx.x;
  const int lane  = tid & 31;
  const int wave  = tid >> 5;
  const int m     = lane & 15;   // M (A) / N (B) index
  const int half  = lane >> 4;   // K-half selector

  const float* __restrict__ xb = x    + (size_t)batch * (LROWS * 3);
  const float* __restrict__ pb = proj + (size_t)batch * (LROWS * 3);

  // One-time constant columns: col 3 = 1.0, cols 4..15 = 0 (both buffers).
#pragma unroll
  for (int b = 0; b < 2; ++b) {
    float* rp = &smem[b][0][tid * PADW];
    float* rx = &smem[b][1][tid * PADW];
#pragma unroll
    for (int c = 3; c < PADW; ++c) {
      const float v = (c == 3) ? 1.0f : 0.0f;
      rp[c] = v;
      rx[c] = v;
    }
  }

  // Coalesced tile fill: thread t owns row t of the tile (3 contiguous floats
  // per array -> global_load_b96), scattered into the padded LDS row.
  auto fill = [&](int buf, int tile) {
    const float* gp = pb + ((size_t)tile * TILE + tid) * 3;
    const float* gx = xb + ((size_t)tile * TILE + tid) * 3;
    const float p0 = gp[0], p1 = gp[1], p2 = gp[2];
    const float x0 = gx[0], x1 = gx[1], x2 = gx[2];
    float* rp = &smem[buf][0][tid * PADW];
    float* rx = &smem[buf][1][tid * PADW];
    rp[0] = p0; rp[1] = p1; rp[2] = p2;
    rx[0] = x0; rx[1] = x1; rx[2] = x2;
  };

  v8f acc = {};

  fill(0, 0);
  for (int tile = 0; tile < NTILES; ++tile) {
    __syncthreads();  // fill(tile) + (first iter) constant cols visible
    if (tile + 1 < NTILES) fill((tile + 1) & 1, tile + 1);  // overlap w/ WMMA
    if (tile + 2 < NTILES) {  // warm L2 for tile+2 (global_prefetch_b8)
      __builtin_prefetch(pb + ((size_t)(tile + 2) * TILE + tid) * 3, 0, 0);
      __builtin_prefetch(xb + ((size_t)(tile + 2) * TILE + tid) * 3, 0, 0);
    }

    const float* tp = &smem[tile & 1][0][0];
    const float* tx = &smem[tile & 1][1][0];
    // Wave w reduces tile rows [w*32, w*32+32): 8 WMMAs of K=4 rows.
    const int rbase = wave * 32 + 2 * half;
#pragma unroll
    for (int s = 0; s < 8; ++s) {
      const int r0 = (rbase + s * 4) * PADW + m;
      v2f a, b;
      a.x = tp[r0];
      a.y = tp[r0 + PADW];
      b.x = tx[r0];
      b.y = tx[r0 + PADW];
      acc = __builtin_amdgcn_wmma_f32_16x16x4_f32(
          /*neg_a=*/false, a, /*neg_b=*/false, b,
          /*c_mod=*/(short)0, acc, /*reuse_a=*/false, /*reuse_b=*/false);
    }
  }

  // D layout: VGPR v, lanes 0-15 -> D[v][lane]. We need D[0..3][0..3]:
  //   D[k][j] = sum proj_k*x_j ; D[3][j] = sum x_j ; D[k][3] = sum proj_k
  if (lane < 4) {
#pragma unroll
    for (int mm = 0; mm < 4; ++mm) part[wave * 16 + mm * 4 + lane] = acc[mm];
  }
  __syncthreads();

  if (tid == 0) {
    float D[4][4];
#pragma unroll
    for (int mm = 0; mm < 4; ++mm)
#pragma unroll
      for (int nn = 0; nn < 4; ++nn) {
        float s = 0.0f;
        for (int w = 0; w < 8; ++w) s += part[w * 16 + mm * 4 + nn];
        D[mm][nn] = s;
      }

    const float invL = 1.0f / (float)LROWS;
    float mu[3];
    for (int j = 0; j < 3; ++j) mu[j] = D[3][j] * invL;

    // init[k][j] = sum proj_k*(x_j - mu_j) = D[k][j] - D[k][3]*mu[j]
    float v[3][3];
    for (int k = 0; k < 3; ++k)
      for (int j = 0; j < 3; ++j) v[k][j] = D[k][j] - D[k][3] * mu[j];

    // row-normalize (init / ||init||, no eps, per reference)
    for (int k = 0; k < 3; ++k) {
      float n = sqrtf(v[k][0]*v[k][0] + v[k][1]*v[k][1] + v[k][2]*v[k][2]);
      float inv = 1.0f / n;
      v[k][0] *= inv; v[k][1] *= inv; v[k][2] *= inv;
    }

    // Gram-Schmidt with eps on denominators
    float u[3][3];
    u[0][0] = v[0][0]; u[0][1] = v[0][1]; u[0][2] = v[0][2];
    float d00 = u[0][0]*u[0][0] + u[0][1]*u[0][1] + u[0][2]*u[0][2] + EPSF;
    float t10 = (v[1][0]*u[0][0] + v[1][1]*u[0][1] + v[1][2]*u[0][2]) / d00;
    for (int j = 0; j < 3; ++j) u[1][j] = v[1][j] - t10 * u[0][j];
    float d11 = u[1][0]*u[1][0] + u[1][1]*u[1][1] + u[1][2]*u[1][2] + EPSF;
    float t20 = (v[2][0]*u[0][0] + v[2][1]*u[0][1] + v[2][2]*u[0][2]) / d00;
    float t21 = (v[2][0]*u[1][0] + v[2][1]*u[1][1] + v[2][2]*u[1][2]) / d11;
    for (int j = 0; j < 3; ++j) u[2][j] = v[2][j] - t20 * u[0][j] - t21 * u[1][j];

    // final normalize: u / (||u|| + eps)
    for (int k = 0; k < 3; ++k) {
      float n = sqrtf(u[k][0]*u[k][0] + u[k][1]*u[k][1] + u[k][2]*u[k][2]) + EPSF;
      float inv = 1.0f / n;
      u[k][0] *= inv; u[k][1] *= inv; u[k][2] *= inv;
    }

    // frame = transpose(gs rows); bias[i] = sum_j u[j][i] * b[j]
    const float* bb = bvec + (size_t)batch * 3;
    const float b0 = bb[0], b1 = bb[1], b2 = bb[2];
    float bias[3];
    for (int i = 0; i < 3; ++i)
      bias[i] = u[0][i]*b0 + u[1][i]*b1 + u[2][i]*b2;

    // softmax over L sums to 1 (identity) -> std = sqrt((1+L*EPS)/L),
    // data-independent; skips two extra full passes over x (~384 MB saved).
    const float stdv  = sqrtf((1.0f + (float)LROWS * EPSF) * invL);
    const float scale = g[batch] / stdv;

    float* pp = params + (size_t)batch * 8;
    pp[0] = mu[0]; pp[1] = mu[1]; pp[2] = mu[2];
    pp[3] = bias[0]; pp[4] = bias[1]; pp[5] = bias[2];
    pp[6] = scale; pp[7] = 0.0f;
  }
}

// ---------------------------------------------------------------------------
// Kernel 2: streaming apply, float4 vectorized. 6144 floats per batch =
// exactly 6 blocks of 1024 floats, so a block never straddles a batch.
// out = scale*(x - mu[comp]) + bias[comp], comp cycles mod 3.
// ---------------------------------------------------------------------------
__global__ __launch_bounds__(256) void norm_apply_kernel(
    const float* __restrict__ x, const float* __restrict__ params,
    float* __restrict__ out) {
  __shared__ float p[8];
  const int batch = blockIdx.x / 6;
  if (threadIdx.x < 8) p[threadIdx.x] = params[(size_t)batch * 8 + threadIdx.x];
  __syncthreads();

  const size_t base = (size_t)blockIdx.x * 1024;
  const float4* __restrict__ xin = (const float4*)(x + base);
  float4* __restrict__ o = (float4*)(out + base);

  float4 v = xin[threadIdx.x];
  // first element's component: (base + 4*tid) % 3 == (blockIdx.x + tid) % 3
  const int c0 = (int)((blockIdx.x + threadIdx.x) % 3u);
  const int cA = (c0 + 1 < 3) ? c0 + 1 : c0 - 2;
  const int cB = (c0 + 2 < 3) ? c0 + 2 : c0 - 1;

  const float s = p[6];
  float4 r;
  r.x = s * (v.x - p[c0]) + p[3 + c0];
  r.y = s * (v.y - p[cA]) + p[3 + cA];
  r.z = s * (v.z - p[cB]) + p[3 + cB];
  r.w = s * (v.w - p[c0]) + p[3 + c0];
  o[threadIdx.x] = r;
}

extern "C" void kernel_launch(void* const* d_in, const int* in_sizes, int n_in,
                              void* d_out, int out_size, void* d_ws, size_t ws_size,
                              hipStream_t stream) {
  const float* x    = (const float*)d_in[0];  // (B*L, 3)
  const float* g    = (const float*)d_in[1];  // (B,)
  const float* b    = (const float*)d_in[2];  // (B, 3)
  const float* proj = (const float*)d_in[3];  // (B*L, 3)
  float* out    = (float*)d_out;
  float* params = (float*)d_ws;               // 8 floats per batch = 256 KB

  norm_reduce_kernel<<<NBATCH, 256, 0, stream>>>(x, g, b, proj, params);
  norm_apply_kernel<<<NBATCH * 6, 256, 0, stream>>>(x, params, out);
}